// BatchTimeSeriesInterpolator_1322849927845
// MI455X (gfx1250) — compile-verified
//
#include <hip/hip_runtime.h>
#include <hip/hip_bf16.h>
#include <stdint.h>

// Harness shapes: times/values (1024, 65536) f32, t (65536,) f32, out (65536,) f32.
#define NT_C           1024      // ntime (power of two, specialized)
#define NB_C           65536     // nbatch (specialized)
#define COLS_PER_BLOCK 256
#define ROWSTRIDE      16        // staged-row stride: rows 16k-1, k=1..63
#define NSTAGED        63        // levels s = 512..16 staged in LDS (63 KB)

typedef unsigned int u32x4 __attribute__((ext_vector_type(4)));
typedef unsigned int u32x8 __attribute__((ext_vector_type(8)));

__device__ __forceinline__ void wait_tensorcnt0() {
#if __has_builtin(__builtin_amdgcn_s_wait_tensorcnt)
    __builtin_amdgcn_s_wait_tensorcnt(0);
#else
    asm volatile("s_wait_tensorcnt 0x0" ::: "memory");
#endif
}

// ---------------------------------------------------------------------------
// Specialized kernel: ntime=1024, nbatch=65536. One TDM descriptor stages the
// 63 shared binary-search rows (16k-1) x 256 cols tile into LDS per block.
// ---------------------------------------------------------------------------
__global__ __launch_bounds__(COLS_PER_BLOCK)
void bts_interp_tdm(const float* __restrict__ times,
                    const float* __restrict__ values,
                    const float* __restrict__ tq,
                    float* __restrict__ out)
{
    __shared__ float smem[NSTAGED * COLS_PER_BLOCK];  // 63 KB, row-major tile

    const int      tid = threadIdx.x;
    const unsigned blk = blockIdx.x;
    const int      col = (int)blk * COLS_PER_BLOCK + tid;

    // ---- Tensor Data Mover: DMA the 63x256 f32 tile global->LDS. ----
    // Tile start = times[row=15, col=blk*256]; row stride = 16 rows.
    if (tid < 32) {  // wave 0 only issues (TDM ignores EXEC; 1 DMA per block)
        const unsigned lds_base = (unsigned)(uintptr_t)(&smem[0]);
        const unsigned long long ga =
            (unsigned long long)times +
            (((unsigned long long)(ROWSTRIDE - 1) * NB_C +
              (unsigned long long)blk * COLS_PER_BLOCK) * 4ull);

        // D# group0 (128b): count=1 | lds_addr | global_addr[56:0] | type=2
        u32x4 g0;
        g0.s0 = 1u;                                   // count=1, user descriptor
        g0.s1 = lds_base;                             // LDS byte address
        g0.s2 = (unsigned)(ga & 0xffffffffull);       // global_addr[31:0]
        g0.s3 = (unsigned)((ga >> 32) & 0x01ffffffull) | 0x80000000u; // +type=2

        // D# group1 (256b): 2D tile descriptor.
        u32x8 g1;
        g1.s0 = 2u << 16;                 // wg_mask=0, data_size=2 (4 bytes)
        g1.s1 = 0u;                       // no atomic barrier; tensor_dim0[15:0]=0
        g1.s2 = (NB_C >> 16)              // tensor_dim0[31:16] = 65536>>16
              | ((NT_C & 0xffffu) << 16); // tensor_dim1[15:0]  = 1024
        g1.s3 = (unsigned)COLS_PER_BLOCK << 16;   // tile_dim0 = 256 elements
        g1.s4 = NSTAGED;                  // tile_dim1 = 63 rows, tile_dim2 = 0
        g1.s5 = (unsigned)ROWSTRIDE * NB_C;  // tensor_dim0_stride = 0x100000
        g1.s6 = 0u;                       // stride[47:32]=0, dim1_stride lo = 0
        g1.s7 = 0u;                       // dim1_stride hi = 0

        asm volatile("tensor_load_to_lds %0, %1"
                     :: "s"(g0), "s"(g1)
                     : "memory");
        wait_tensorcnt0();                // drain TENSORcnt on issuing wave
    }
    __syncthreads();                      // publish LDS tile to all 8 waves

    const float t = tq[col];

    // ---- Branchless upper-bound: c = #{ times[i,col] <= t }. Track the last
    //      success/fail probe values so t0=times[c-1], t1=times[c] are free. ----
    unsigned c = 0;
    float t0 = -3.402823466e38f;
    float t1 =  3.402823466e38f;

    // Levels s = 512..16 from LDS (bank = tid mod 64 -> conflict-free).
    #pragma unroll
    for (int s = NT_C / 2; s >= ROWSTRIDE; s >>= 1) {
        const unsigned k = (c + (unsigned)s) / (unsigned)ROWSTRIDE - 1u;
        const float x = smem[k * COLS_PER_BLOCK + (unsigned)tid];
        const bool le = (x <= t);
        c  = le ? c + (unsigned)s : c;
        t0 = le ? x : t0;
        t1 = le ? t1 : x;
    }

    // Levels s = 8..1: per-lane gathers; index math is a shift (NB_C = 2^16).
    const float* colT = times + col;
    #pragma unroll
    for (int s = ROWSTRIDE / 2; s >= 1; s >>= 1) {
        const float x = colT[(size_t)(c + (unsigned)s - 1u) << 16];
        const bool le = (x <= t);
        c  = le ? c + (unsigned)s : c;
        t0 = le ? x : t0;
        t1 = le ? t1 : x;
    }

    // ---- Epilogue: gi = c mod 1024, torch negative-index wrap semantics. ----
    const float* colV = values + col;
    float r;
    if (c == 0u || c == (unsigned)NT_C) {
        // t outside covered range: last knot + last slope (wrap case).
        const float ta = colT[(size_t)(NT_C - 2) << 16];
        const float tb = colT[(size_t)(NT_C - 1) << 16];
        const float va = colV[(size_t)(NT_C - 2) << 16];
        const float vb = colV[(size_t)(NT_C - 1) << 16];
        r = vb + (vb - va) / (tb - ta) * (t - tb);
    } else {
        const float v0 = colV[(size_t)(c - 1u) << 16];
        const float v1 = colV[(size_t)c        << 16];
        r = v0 + (v1 - v0) / (t1 - t0) * (t - t0);
    }
    out[col] = r;
}

// ---------------------------------------------------------------------------
// Generic fallback (any ntime/nbatch): plain per-lane binary search.
// ---------------------------------------------------------------------------
__global__ void bts_interp_generic(const float* __restrict__ times,
                                   const float* __restrict__ values,
                                   const float* __restrict__ tq,
                                   float* __restrict__ out,
                                   int ntime, int nbatch)
{
    const int b = blockIdx.x * blockDim.x + threadIdx.x;
    if (b >= nbatch) return;
    const float t = tq[b];

    int lo = 0, hi = ntime;                 // c = #elements <= t
    while (lo < hi) {
        const int mid = (lo + hi) >> 1;
        const float x = times[(size_t)mid * nbatch + b];
        if (x <= t) lo = mid + 1; else hi = mid;
    }
    const int c = lo;

    const float* colT = times + b;
    const float* colV = values + b;
    float r;
    if (c == 0 || c == ntime) {
        const float ta = colT[(size_t)(ntime - 2) * nbatch];
        const float tb = colT[(size_t)(ntime - 1) * nbatch];
        const float va = colV[(size_t)(ntime - 2) * nbatch];
        const float vb = colV[(size_t)(ntime - 1) * nbatch];
        r = vb + (vb - va) / (tb - ta) * (t - tb);
    } else {
        const float t0 = colT[(size_t)(c - 1) * nbatch];
        const float t1 = colT[(size_t)c * nbatch];
        const float v0 = colV[(size_t)(c - 1) * nbatch];
        const float v1 = colV[(size_t)c * nbatch];
        r = v0 + (v1 - v0) / (t1 - t0) * (t - t0);
    }
    out[b] = r;
}

extern "C" void kernel_launch(void* const* d_in, const int* in_sizes, int n_in,
                              void* d_out, int out_size, void* d_ws, size_t ws_size,
                              hipStream_t stream) {
    const float* times  = (const float*)d_in[0];
    const float* values = (const float*)d_in[1];
    const float* tq     = (const float*)d_in[2];
    float* out = (float*)d_out;

    const int nbatch = in_sizes[2];
    const int ntime  = in_sizes[0] / nbatch;

    if (ntime == NT_C && nbatch == NB_C) {
        bts_interp_tdm<<<NB_C / COLS_PER_BLOCK, COLS_PER_BLOCK, 0, stream>>>(
            times, values, tq, out);
    } else {
        const int threads = 256;
        const int blocks  = (nbatch + threads - 1) / threads;
        bts_interp_generic<<<blocks, threads, 0, stream>>>(
            times, values, tq, out, ntime, nbatch);
    }
}